// LinearAttention_13520557048071
// MI455X (gfx1250) — compile-verified
//
#include <hip/hip_runtime.h>
#include <math.h>

#define NN 4096
#define LL 32
#define DD 300
#define MM 300
#define MP 304            // 300 padded to multiple of 16 (19 tiles)
#define NEGV (-1e30f)

typedef __attribute__((ext_vector_type(2))) float v2f;
typedef __attribute__((ext_vector_type(8))) float v8f;

#define V8F_ZERO {0.f, 0.f, 0.f, 0.f, 0.f, 0.f, 0.f, 0.f}

__device__ __forceinline__ v8f wmma_f32(v2f a, v2f b, v8f c) {
    // D = A(16x4) * B(4x16) + C, fp32 exact
    return __builtin_amdgcn_wmma_f32_16x16x4_f32(
        /*neg_a=*/false, a, /*neg_b=*/false, b,
        /*c_mod=*/(short)0, c, /*reuse_a=*/false, /*reuse_b=*/false);
}

// ---------------------------------------------------------------------------
// Kernel 0: pad aspect_v [N,300] -> AV [N,304]; pad W [300,300] -> Wp [304,300]
// with zero rows; pad bias -> bp [304].  Removes all masking ALU from the
// WMMA inner loops.
// ---------------------------------------------------------------------------
__global__ void pad_av_kernel(const float* __restrict__ av, float* __restrict__ AV) {
    int i = blockIdx.x * blockDim.x + threadIdx.x;
    if (i >= NN * MP) return;
    int r = i / MP, c = i % MP;
    AV[i] = (c < DD) ? av[r * DD + c] : 0.0f;
}

__global__ void pad_wb_kernel(const float* __restrict__ W, const float* __restrict__ b,
                              float* __restrict__ Wp, float* __restrict__ bp) {
    int i = blockIdx.x * blockDim.x + threadIdx.x;
    if (i < MP * DD) {
        int m = i / DD;                 // [MP, DD] and [MM, DD] share row-major index
        Wp[i] = (m < MM) ? W[i] : 0.0f;
    }
    if (i < MP) bp[i] = (i < MM) ? b[i] : 0.0f;
}

// ---------------------------------------------------------------------------
// Kernel 2: Q_raw = aspect_v @ Wp^T + bp -> [N, MP] (pad cols naturally 0)
// 2x1 register blocking: each wave does a 32x16 region, B operand shared.
// ---------------------------------------------------------------------------
__global__ void qgemm_kernel(const float* __restrict__ X,   // [N, D]
                             const float* __restrict__ Wp,  // [MP, D]
                             const float* __restrict__ bp,  // [MP]
                             float* __restrict__ Out)       // [N, MP]
{
    int wave = (blockIdx.x * blockDim.x + threadIdx.x) >> 5;
    int lane = threadIdx.x & 31;
    const int ntc = MP / 16;                    // 19 col tiles
    const int tiles = (NN / 32) * ntc;          // 2432
    if (wave >= tiles) return;
    int ti = wave / ntc, tc = wave % ntc;
    int i0 = ti * 32, c0 = tc * 16;
    int lr = lane & 15, half = lane >> 4;

    const float* a0 = X + (size_t)(i0 + lr) * DD + 2 * half;
    const float* a1 = a0 + (size_t)16 * DD;
    const float* bb = Wp + (size_t)(c0 + lr) * DD + 2 * half;

    v8f acc0 = V8F_ZERO, acc1 = V8F_ZERO;
    for (int k = 0; k < DD; k += 4) {
        v2f b  = *(const v2f*)(bb + k);
        v2f va = *(const v2f*)(a0 + k);
        v2f vb = *(const v2f*)(a1 + k);
        acc0 = wmma_f32(va, b, acc0);
        acc1 = wmma_f32(vb, b, acc1);
    }
    float bv = bp[c0 + lr];
#pragma unroll
    for (int e = 0; e < 8; ++e) {
        int r = i0 + e + 8 * half;
        Out[(size_t)r * MP + c0 + lr]        = acc0[e] + bv;
        Out[(size_t)(r + 16) * MP + c0 + lr] = acc1[e] + bv;
    }
}

// ---------------------------------------------------------------------------
// Kernel 3: row-wise l2 normalize Q in place ([N, MP], pad cols are zero)
// ---------------------------------------------------------------------------
__global__ void rownorm_kernel(float* __restrict__ Q) {
    __shared__ float red[8];
    int row = blockIdx.x;
    float* p = Q + (size_t)row * MP;
    int tid = threadIdx.x;
    float ss = 0.f;
    for (int c = tid; c < MP; c += 256) { float v = p[c]; ss += v * v; }
#pragma unroll
    for (int o = 16; o > 0; o >>= 1) ss += __shfl_xor(ss, o, 32);
    if ((tid & 31) == 0) red[tid >> 5] = ss;
    __syncthreads();
    if (tid == 0) {
        float t = 0.f;
        for (int w = 0; w < 8; ++w) t += red[w];
        red[0] = 1.0f / fmaxf(sqrtf(t), 1e-12f);
    }
    __syncthreads();
    float inv = red[0];
    for (int c = tid; c < MP; c += 256) p[c] *= inv;
}

// ---------------------------------------------------------------------------
// Kernel 4 (fused): per n: Fn = feature[n] @ Wp^T + bp (32 x MP in LDS),
// l2norm each of the 32 rows, mean over rows -> Fm[n, :]
// 8 waves share 19 col tiles; each wave computes both 16-row tiles per col
// tile, sharing the B operand.
// ---------------------------------------------------------------------------
__global__ void featgemm_kernel(const float* __restrict__ F,   // [N, L, D]
                                const float* __restrict__ Wp,  // [MP, D]
                                const float* __restrict__ bp,  // [MP]
                                float* __restrict__ Fm)        // [N, MP]
{
    __shared__ float tile[LL][MP];   // 38,912 bytes
    __shared__ float inv[LL];
    int n = blockIdx.x;
    const float* X = F + (size_t)n * LL * DD;   // [32, 300]
    int wave = threadIdx.x >> 5, lane = threadIdx.x & 31;
    int lr = lane & 15, half = lane >> 4;
    const int ntc = MP / 16;                    // 19

    const float* a0 = X + (size_t)lr * DD + 2 * half;        // rows 0..15
    const float* a1 = X + (size_t)(16 + lr) * DD + 2 * half; // rows 16..31

    for (int c = wave; c < ntc; c += 8) {
        int c0 = c * 16;
        const float* bb = Wp + (size_t)(c0 + lr) * DD + 2 * half;
        v8f acc0 = V8F_ZERO, acc1 = V8F_ZERO;
        for (int k = 0; k < DD; k += 4) {
            v2f b  = *(const v2f*)(bb + k);
            v2f va = *(const v2f*)(a0 + k);
            v2f vb = *(const v2f*)(a1 + k);
            acc0 = wmma_f32(va, b, acc0);
            acc1 = wmma_f32(vb, b, acc1);
        }
        float bv = bp[c0 + lr];
#pragma unroll
        for (int e = 0; e < 8; ++e) {
            tile[e + 8 * half][c0 + lr]      = acc0[e] + bv;
            tile[16 + e + 8 * half][c0 + lr] = acc1[e] + bv;
        }
    }
    __syncthreads();

    if (threadIdx.x < LL) {
        float ss = 0.f;
        for (int c = 0; c < MP; ++c) { float v = tile[threadIdx.x][c]; ss += v * v; }
        inv[threadIdx.x] = 1.0f / fmaxf(sqrtf(ss), 1e-12f);
    }
    __syncthreads();

    for (int c = threadIdx.x; c < MP; c += 256) {
        float s = 0.f;
#pragma unroll 8
        for (int r = 0; r < LL; ++r) s += tile[r][c] * inv[r];
        Fm[(size_t)n * MP + c] = s * (1.0f / LL);
    }
}

// ---------------------------------------------------------------------------
// Kernel 5: S = (Q@Q^T + Fm@Fm^T) * dmask ; zero -> NEG.
// 2x2 register blocking: each wave computes a 32x32 region (4 tiles),
// 4 WMMAs per 4 b64 loads -> 2x the FLOP/byte of the unblocked version.
// ---------------------------------------------------------------------------
__global__ void logits_kernel(const float* __restrict__ Q,     // [N, MP]
                              const float* __restrict__ Fm,    // [N, MP]
                              const float* __restrict__ dmask, // [N, N]
                              float* __restrict__ S)           // [N, N]
{
    int wave = (blockIdx.x * blockDim.x + threadIdx.x) >> 5;
    int lane = threadIdx.x & 31;
    int lr = lane & 15, half = lane >> 4;
    const int nt2 = NN / 32;                    // 128
    int ti = wave / nt2, tj = wave % nt2;
    int i0 = ti * 32, j0 = tj * 32;

    v8f acc00 = V8F_ZERO, acc01 = V8F_ZERO, acc10 = V8F_ZERO, acc11 = V8F_ZERO;

    {
        const float* a0 = Q + (size_t)(i0 + lr) * MP + 2 * half;
        const float* a1 = a0 + (size_t)16 * MP;
        const float* b0 = Q + (size_t)(j0 + lr) * MP + 2 * half;
        const float* b1 = b0 + (size_t)16 * MP;
        for (int k = 0; k < MP; k += 4) {
            v2f va0 = *(const v2f*)(a0 + k);
            v2f va1 = *(const v2f*)(a1 + k);
            v2f vb0 = *(const v2f*)(b0 + k);
            v2f vb1 = *(const v2f*)(b1 + k);
            acc00 = wmma_f32(va0, vb0, acc00);
            acc01 = wmma_f32(va0, vb1, acc01);
            acc10 = wmma_f32(va1, vb0, acc10);
            acc11 = wmma_f32(va1, vb1, acc11);
        }
    }
    {
        const float* a0 = Fm + (size_t)(i0 + lr) * MP + 2 * half;
        const float* a1 = a0 + (size_t)16 * MP;
        const float* b0 = Fm + (size_t)(j0 + lr) * MP + 2 * half;
        const float* b1 = b0 + (size_t)16 * MP;
        for (int k = 0; k < MP; k += 4) {
            v2f va0 = *(const v2f*)(a0 + k);
            v2f va1 = *(const v2f*)(a1 + k);
            v2f vb0 = *(const v2f*)(b0 + k);
            v2f vb1 = *(const v2f*)(b1 + k);
            acc00 = wmma_f32(va0, vb0, acc00);
            acc01 = wmma_f32(va0, vb1, acc01);
            acc10 = wmma_f32(va1, vb0, acc10);
            acc11 = wmma_f32(va1, vb1, acc11);
        }
    }

#pragma unroll
    for (int e = 0; e < 8; ++e) {
        int r0 = i0 + e + 8 * half;
        size_t i00 = (size_t)r0 * NN + j0 + lr;
        size_t i01 = i00 + 16;
        size_t i10 = i00 + (size_t)16 * NN;
        size_t i11 = i10 + 16;
        float v;
        v = acc00[e] * dmask[i00]; S[i00] = (v == 0.0f) ? NEGV : v;
        v = acc01[e] * dmask[i01]; S[i01] = (v == 0.0f) ? NEGV : v;
        v = acc10[e] * dmask[i10]; S[i10] = (v == 0.0f) ? NEGV : v;
        v = acc11[e] * dmask[i11]; S[i11] = (v == 0.0f) ? NEGV : v;
    }
}

// ---------------------------------------------------------------------------
// Kernel 6: row-wise softmax of S in place (one block per row)
// ---------------------------------------------------------------------------
__global__ void softmax_kernel(float* __restrict__ S) {
    __shared__ float red[8];
    int row = blockIdx.x;
    float* p = S + (size_t)row * NN;
    int tid = threadIdx.x;

    float mx = -3.4e38f;
    for (int c = tid; c < NN; c += 256) mx = fmaxf(mx, p[c]);
#pragma unroll
    for (int o = 16; o > 0; o >>= 1) mx = fmaxf(mx, __shfl_xor(mx, o, 32));
    if ((tid & 31) == 0) red[tid >> 5] = mx;
    __syncthreads();
    if (tid == 0) {
        float t = red[0];
        for (int w = 1; w < 8; ++w) t = fmaxf(t, red[w]);
        red[0] = t;
    }
    __syncthreads();
    mx = red[0];
    __syncthreads();

    float sum = 0.f;
    for (int c = tid; c < NN; c += 256) {
        float e = __expf(p[c] - mx);
        p[c] = e;
        sum += e;
    }
#pragma unroll
    for (int o = 16; o > 0; o >>= 1) sum += __shfl_xor(sum, o, 32);
    if ((tid & 31) == 0) red[tid >> 5] = sum;
    __syncthreads();
    if (tid == 0) {
        float t = 0.f;
        for (int w = 0; w < 8; ++w) t += red[w];
        red[0] = 1.0f / t;
    }
    __syncthreads();
    float inv = red[0];
    for (int c = tid; c < NN; c += 256) p[c] *= inv;
}

// ---------------------------------------------------------------------------
// Kernel 7: out[N, 300] = S[N,N] @ AV[N, MP]
// 2x1 row blocking: 32x16 region per wave, B operand shared across row tiles.
// ---------------------------------------------------------------------------
__global__ void outgemm_kernel(const float* __restrict__ S,   // [N, N]
                               const float* __restrict__ AV,  // [N, MP] padded
                               float* __restrict__ Out)       // [N, D]
{
    int wave = (blockIdx.x * blockDim.x + threadIdx.x) >> 5;
    int lane = threadIdx.x & 31;
    const int ntc = MP / 16;                    // 19
    const int tiles = (NN / 32) * ntc;          // 2432
    if (wave >= tiles) return;
    int ti = wave / ntc, tc = wave % ntc;
    int i0 = ti * 32, d0 = tc * 16;
    int lr = lane & 15, half = lane >> 4;

    const float* a0 = S + (size_t)(i0 + lr) * NN + 2 * half;
    const float* a1 = a0 + (size_t)16 * NN;
    const float* bbase = AV + (size_t)(2 * half) * MP + d0 + lr;

    v8f acc0 = V8F_ZERO, acc1 = V8F_ZERO;
    for (int k = 0; k < NN; k += 4) {
        const float* bp = bbase + (size_t)k * MP;
        v2f b  = {bp[0], bp[MP]};
        v2f va = *(const v2f*)(a0 + k);
        v2f vb = *(const v2f*)(a1 + k);
        acc0 = wmma_f32(va, b, acc0);
        acc1 = wmma_f32(vb, b, acc1);
    }
    int col = d0 + lr;
    if (col < DD) {
#pragma unroll
        for (int e = 0; e < 8; ++e) {
            int r = i0 + e + 8 * half;
            Out[(size_t)r * DD + col]        = acc0[e];
            Out[(size_t)(r + 16) * DD + col] = acc1[e];
        }
    }
}

// ---------------------------------------------------------------------------
// launch
// ---------------------------------------------------------------------------
extern "C" void kernel_launch(void* const* d_in, const int* in_sizes, int n_in,
                              void* d_out, int out_size, void* d_ws, size_t ws_size,
                              hipStream_t stream) {
    const float* feature = (const float*)d_in[0];   // [N, L, D]
    const float* aspect  = (const float*)d_in[1];   // [N, D]
    const float* dmask   = (const float*)d_in[2];   // [N, N]
    const float* W       = (const float*)d_in[3];   // [M, D]
    const float* b       = (const float*)d_in[4];   // [M]
    float* out = (float*)d_out;                     // [N, D]

    float* ws = (float*)d_ws;
    const size_t NMP = (size_t)NN * MP;
    float* Q  = ws;                         // [N, MP]
    float* Fm = ws + NMP;                   // [N, MP]
    float* AV = ws + 2 * NMP;               // [N, MP]
    float* Wp = ws + 3 * NMP;               // [MP, D]
    float* bp = Wp + (size_t)MP * DD;       // [MP]
    float* S  = bp + MP;                    // [N, N]

    const int gemm_tiles = (NN / 32) * (MP / 16);         // 2432
    const int gemm_blocks = (gemm_tiles + 7) / 8;         // 304
    const int logit_blocks = (NN / 32) * (NN / 32) / 8;   // 16384 waves / 8

    pad_av_kernel<<<(NN * MP + 255) / 256, 256, 0, stream>>>(aspect, AV);
    pad_wb_kernel<<<(MP * DD + 255) / 256, 256, 0, stream>>>(W, b, Wp, bp);
    qgemm_kernel<<<gemm_blocks, 256, 0, stream>>>(aspect, Wp, bp, Q);
    rownorm_kernel<<<NN, 256, 0, stream>>>(Q);
    featgemm_kernel<<<NN, 256, 0, stream>>>(feature, Wp, bp, Fm);
    logits_kernel<<<logit_blocks, 256, 0, stream>>>(Q, Fm, dmask, S);
    softmax_kernel<<<NN, 256, 0, stream>>>(S);
    outgemm_kernel<<<gemm_blocks, 256, 0, stream>>>(S, AV, out);
}